// EmergencyGNNSimple_72112500900411
// MI455X (gfx1250) — compile-verified
//
#include <hip/hip_runtime.h>
#include <hip/hip_bf16.h>

// ---------------------------------------------------------------------------
// GNN link predictor for MI455X (gfx1250, wave32, WMMA).
//   N=100000 nodes (=6250*16, tiles exact), E=3.2M edges, Q=500K queries.
//   Dense linears -> V_WMMA_F32_16X16X4_F32 (full f32 precision, tiny FLOPs).
//   Message passing -> L2-resident gather + global_atomic_add_f32 scatter.
// ---------------------------------------------------------------------------

typedef float v2f __attribute__((ext_vector_type(2)));
typedef float v8f __attribute__((ext_vector_type(8)));

#define NODE_DIM 32
#define HIDDEN   64

__device__ __forceinline__ void atomic_add_f32(float* p, float v) {
    unsafeAtomicAdd(p, v);   // lowers to global_atomic_add_f32 on gfx1250
}

// ---------------- degree / normalization ----------------------------------
__global__ void count_deg_kernel(const long long* __restrict__ dst,
                                 float* __restrict__ deg, int E) {
    int i = blockIdx.x * blockDim.x + threadIdx.x;
    if (i < E) atomic_add_f32(&deg[dst[i]], 1.0f);
}

__global__ void inv_sqrt_kernel(float* __restrict__ deg, int N) {
    int i = blockIdx.x * blockDim.x + threadIdx.x;
    if (i < N) deg[i] = __frsqrt_rn(deg[i] + 1.0f);  // includes self-loop
}

// ---------------- WMMA dense linear: out = [relu](A @ W [+ b]) -------------
// A: [M, K] row-major, W: [K, NOUT] row-major, out: [M, NOUT].
// One wave computes one 16x16 output tile via K/4 chained f32 WMMAs.
// 32-bit A 16x4 layout: lanes 0-15 hold K=k,k+1; lanes 16-31 hold K=k+2,k+3.
// 32-bit B 4x16 layout: row striped across lanes; VGPR0/1 = rows k,k+1
// (lanes 0-15) and k+2,k+3 (lanes 16-31).
// C/D 16x16: VGPR v holds row v (lanes 0-15) and row v+8 (lanes 16-31).
template <int K, int NOUT, bool BIAS, bool RELU>
__global__ void linear_wmma_kernel(const float* __restrict__ A,
                                   const float* __restrict__ W,
                                   const float* __restrict__ bias,
                                   float* __restrict__ out, int M) {
    const int tilesN = NOUT / 16;
    const int totalTiles = (M / 16) * tilesN;
    int wave = threadIdx.x >> 5;
    int lane = threadIdx.x & 31;
    int tile = blockIdx.x * 8 + wave;           // wave-uniform guard -> EXEC all 1s
    if (tile >= totalTiles) return;
    int tm = (tile / tilesN) * 16;
    int tn = (tile % tilesN) * 16;

    int half = lane >> 4;                       // 0: lanes 0-15, 1: lanes 16-31
    const float* arow = A + (size_t)(tm + (lane & 15)) * K;
    const float* wcol = W + tn + (lane & 15);

    v8f c = {};
#pragma unroll
    for (int k = 0; k < K; k += 4) {
        int kk = k + (half << 1);
        v2f a, b;
        a.x = arow[kk];
        a.y = arow[kk + 1];
        b.x = wcol[(size_t)kk * NOUT];
        b.y = wcol[(size_t)(kk + 1) * NOUT];
        c = __builtin_amdgcn_wmma_f32_16x16x4_f32(
                /*neg_a=*/false, a, /*neg_b=*/false, b,
                /*c_mod=*/(short)0, c, /*reuse_a=*/false, /*reuse_b=*/false);
    }

    int col   = tn + (lane & 15);
    int rbase = tm + (half << 3);
    float bv  = BIAS ? bias[col] : 0.0f;
#pragma unroll
    for (int v = 0; v < 8; ++v) {
        float val = c[v] + bv;
        if (RELU) val = fmaxf(val, 0.0f);
        out[(size_t)(rbase + v) * NOUT + col] = val;
    }
}

// ---------------- edge message scatter -------------------------------------
// element t -> edge e = t/F, feature f = t%F. Lanes contiguous in f =>
// coalesced gather of h[src] rows (L2-resident), f32 atomic scatter to agg.
template <int F>
__global__ void scatter_msg_kernel(const float* __restrict__ h,
                                   const long long* __restrict__ src,
                                   const long long* __restrict__ dst,
                                   const float* __restrict__ inv,
                                   float* __restrict__ agg, int E) {
    int e = blockIdx.x * (256 / F) + (threadIdx.x / F);
    int f = threadIdx.x & (F - 1);
    if (e >= E) return;
    long long s = src[e], d = dst[e];
    float coef = inv[s] * inv[d];
    atomic_add_f32(&agg[(size_t)d * F + f], h[(size_t)s * F + f] * coef);
}

// ---------------- self-loop + bias (+relu) ---------------------------------
template <int F, bool RELU>
__global__ void selfloop_bias_kernel(const float* __restrict__ agg,
                                     const float* __restrict__ t,
                                     const float* __restrict__ inv,
                                     const float* __restrict__ b,
                                     float* __restrict__ out, int N) {
    size_t idx = (size_t)blockIdx.x * blockDim.x + threadIdx.x;
    int i = (int)(idx / F);
    int f = (int)(idx % F);
    if (i >= N) return;
    float ii = inv[i];
    float v = agg[idx] + t[idx] * ii * ii + b[f];
    if (RELU) v = fmaxf(v, 0.0f);
    out[idx] = v;
}

// ---------------- edge predictor MLP ---------------------------------------
// ef = h[a] + h[b] (32f); z = relu(ef@W1 + b1) (16f); sigmoid(z@W2 + b2).
__global__ void predictor_kernel(const float* __restrict__ h,
                                 const long long* __restrict__ qa,
                                 const long long* __restrict__ qb,
                                 const float* __restrict__ W1,
                                 const float* __restrict__ b1,
                                 const float* __restrict__ W2,
                                 const float* __restrict__ b2,
                                 float* __restrict__ out, int Q) {
    __shared__ float sW1[32 * 16];
    __shared__ float sb1[16];
    __shared__ float sW2[16];
    __shared__ float sb2;
    for (int i = threadIdx.x; i < 512; i += blockDim.x) sW1[i] = W1[i];
    if (threadIdx.x < 16) { sb1[threadIdx.x] = b1[threadIdx.x]; sW2[threadIdx.x] = W2[threadIdx.x]; }
    if (threadIdx.x == 0) sb2 = b2[0];
    __syncthreads();

    int q = blockIdx.x * blockDim.x + threadIdx.x;
    if (q >= Q) return;
    long long ia = qa[q], ib = qb[q];
    const float4* pa = (const float4*)(h + (size_t)ia * 32);
    const float4* pb = (const float4*)(h + (size_t)ib * 32);
    float ef[32];
#pragma unroll
    for (int v = 0; v < 8; ++v) {
        float4 x = pa[v], y = pb[v];
        ef[4 * v + 0] = x.x + y.x;
        ef[4 * v + 1] = x.y + y.y;
        ef[4 * v + 2] = x.z + y.z;
        ef[4 * v + 3] = x.w + y.w;
    }
    float acc = sb2;
#pragma unroll
    for (int j = 0; j < 16; ++j) {
        float z = sb1[j];
#pragma unroll
        for (int i = 0; i < 32; ++i) z = fmaf(ef[i], sW1[i * 16 + j], z);
        acc = fmaf(fmaxf(z, 0.0f), sW2[j], acc);
    }
    out[q] = 1.0f / (1.0f + __expf(-acc));
}

// ---------------------------------------------------------------------------
extern "C" void kernel_launch(void* const* d_in, const int* in_sizes, int n_in,
                              void* d_out, int out_size, void* d_ws, size_t ws_size,
                              hipStream_t stream) {
    const float*     x     = (const float*)d_in[0];
    const long long* ei    = (const long long*)d_in[1];   // int64 [2, E]
    const long long* eli   = (const long long*)d_in[2];   // int64 [2, Q]
    const float*     enc_W = (const float*)d_in[3];
    const float*     enc_b = (const float*)d_in[4];
    const float*     c1_W  = (const float*)d_in[5];
    const float*     c1_b  = (const float*)d_in[6];
    const float*     c2_W  = (const float*)d_in[7];
    const float*     c2_b  = (const float*)d_in[8];
    const float*     p1_W  = (const float*)d_in[9];
    const float*     p1_b  = (const float*)d_in[10];
    const float*     p2_W  = (const float*)d_in[11];
    const float*     p2_b  = (const float*)d_in[12];
    float* out = (float*)d_out;

    const int N = in_sizes[0] / NODE_DIM;    // 100000
    const int E = in_sizes[1] / 2;           // 3200000
    const int Q = in_sizes[2] / 2;           // 500000
    const long long* src  = ei;
    const long long* dst  = ei + E;
    const long long* qsrc = eli;
    const long long* qdst = eli + Q;

    // Workspace layout (floats): deg/inv [N] | A [N*64] | B [N*64] | C [N*64]
    float* inv = (float*)d_ws;
    float* A   = inv + N;                    // h1 -> h2 -> h3
    float* B   = A + (size_t)N * HIDDEN;     // t1 -> agg2
    float* C   = B + (size_t)N * HIDDEN;     // agg1 -> t2

    // Re-zero accumulators every call (harness does not re-poison ws).
    hipMemsetAsync(inv, 0, (size_t)N * sizeof(float), stream);
    hipMemsetAsync(C,   0, (size_t)N * HIDDEN * sizeof(float), stream);

    // deg + symmetric normalization
    count_deg_kernel<<<(E + 255) / 256, 256, 0, stream>>>(dst, inv, E);
    inv_sqrt_kernel<<<(N + 255) / 256, 256, 0, stream>>>(inv, N);

    const int tilesEnc = (N / 16) * (HIDDEN / 16);      // 25000 tiles
    const int tilesC2  = (N / 16) * ((HIDDEN / 2) / 16);

    // h1 = relu(x @ enc_W + enc_b)                      [N,64] -> A
    linear_wmma_kernel<NODE_DIM, HIDDEN, true, true>
        <<<(tilesEnc + 7) / 8, 256, 0, stream>>>(x, enc_W, enc_b, A, N);
    // t1 = h1 @ c1_W                                    [N,64] -> B
    linear_wmma_kernel<HIDDEN, HIDDEN, false, false>
        <<<(tilesEnc + 7) / 8, 256, 0, stream>>>(A, c1_W, nullptr, B, N);
    // agg1 += coef * t1[src]                            [N,64] -> C
    scatter_msg_kernel<HIDDEN>
        <<<((size_t)E * HIDDEN + 255) / 256, 256, 0, stream>>>(B, src, dst, inv, C, E);
    // h2 = relu(agg1 + t1*inv^2 + c1_b)                 [N,64] -> A (h1 dead)
    selfloop_bias_kernel<HIDDEN, true>
        <<<((size_t)N * HIDDEN + 255) / 256, 256, 0, stream>>>(C, B, inv, c1_b, A, N);

    // agg2 buffer = front half of B (t1 dead after h2)
    hipMemsetAsync(B, 0, (size_t)N * (HIDDEN / 2) * sizeof(float), stream);

    // t2 = h2 @ c2_W                                    [N,32] -> C (agg1 dead)
    linear_wmma_kernel<HIDDEN, HIDDEN / 2, false, false>
        <<<(tilesC2 + 7) / 8, 256, 0, stream>>>(A, c2_W, nullptr, C, N);
    // agg2 += coef * t2[src]                            [N,32] -> B
    scatter_msg_kernel<HIDDEN / 2>
        <<<((size_t)E * (HIDDEN / 2) + 255) / 256, 256, 0, stream>>>(C, src, dst, inv, B, E);
    // h3 = agg2 + t2*inv^2 + c2_b                       [N,32] -> A (h2 dead)
    selfloop_bias_kernel<HIDDEN / 2, false>
        <<<((size_t)N * (HIDDEN / 2) + 255) / 256, 256, 0, stream>>>(B, C, inv, c2_b, A, N);

    // edge predictor MLP + sigmoid                      [Q] -> out
    predictor_kernel<<<(Q + 255) / 256, 256, 0, stream>>>(
        A, qsrc, qdst, p1_W, p1_b, p2_W, p2_b, out, Q);
}